// GIN_71038759076227
// MI455X (gfx1250) — compile-verified
//
#include <hip/hip_runtime.h>
#include <hip/hip_bf16.h>

// ---------------------------------------------------------------------------
// GIN forward for MI455X (gfx1250, wave32).
//  - fp32 WMMA (v_wmma_f32_16x16x4_f32) for the node MLP GEMMs (exact fp32).
//  - W1/W2 pre-transposed into LDS (stride 68) so every B fragment is one
//    bank-conflict-free ds_load_b64 into an even-aligned VGPR pair (no mov
//    packing before WMMA).
//  - L2-resident scatter-add (global_atomic_add_f32): 25.6MB agg << 192MB L2.
//  - BN statistics accumulated with global_atomic_add_f64 for precision.
// Input pointer order (JAX pytree flatten, dict keys sorted):
//   0:x 1:edge_index 2:batch
//   3+7l .. : conv l -> W1, W2, b1, b2, beta, eps, gamma
//   24:lin1_W 25:lin1_b 26:lin2_W 27:lin2_b
// ---------------------------------------------------------------------------

#define N_NODES_C    100000
#define N_EDGES_C    1600000
#define HID          64
#define WT_STRIDE    68      // padded LDS row stride (floats), conflict-free
#define NUM_GRAPHS_C 256
#define NUM_CLASSES  10

typedef __attribute__((ext_vector_type(2))) float v2f;
typedef __attribute__((ext_vector_type(8))) float v8f;

// ----------------------------- utility kernels -----------------------------

__global__ void k_zero(float* __restrict__ p, int n) {
  int g = blockIdx.x * blockDim.x + threadIdx.x;
  if (g < n) p[g] = 0.f;
}

// one thread = 4 features of one edge; 4 f32 atomics into L2-resident agg
__global__ void k_scatter(const float* __restrict__ x,
                          const int* __restrict__ src,
                          const int* __restrict__ dst,
                          float* __restrict__ agg, int nEdges) {
  int g = blockIdx.x * blockDim.x + threadIdx.x;
  if (g >= nEdges * 16) return;
  int e  = g >> 4;
  int f4 = (g & 15) * 4;
  int s = src[e], d = dst[e];
  const float4 v = *(const float4*)(x + (size_t)s * HID + f4);
  float* p = agg + (size_t)d * HID + f4;
  atomicAdd(p + 0, v.x);
  atomicAdd(p + 1, v.y);
  atomicAdd(p + 2, v.z);
  atomicAdd(p + 3, v.w);
}

// --------------------------- fused GIN MLP (WMMA) ---------------------------
// 128 threads = 4 waves; each wave computes 16 node rows through both GEMMs.
// A-frag layout (f32 16x16x4): lane M = lane&15, K-pair = (lane>>4)*2.
// B-frag: lane N = lane&15, same K-pair -> Wt[n][k..k+1] = one ds_load_b64.
// C/D layout: vgpr r -> M = r (lanes 0-15) / r+8 (lanes 16-31), N = lane&15.
__global__ __launch_bounds__(128) void k_mlp(
    const float* __restrict__ xin, const float* __restrict__ agg,
    const float* __restrict__ W1, const float* __restrict__ bias1,
    const float* __restrict__ W2, const float* __restrict__ bias2,
    const float* __restrict__ epsp,
    float* __restrict__ hpre, double* __restrict__ stats, int nNodes) {
  __shared__ float W1t[HID * WT_STRIDE];   // [n][k] transposed, padded
  __shared__ float W2t[HID * WT_STRIDE];
  __shared__ float sb1[HID], sb2[HID];
  __shared__ float ssum[HID], ssqs[HID];
  __shared__ float tileT[4][16 * WT_STRIDE];  // per-wave C->A transpose

  const int tid = threadIdx.x;
  // load + transpose weights: Wt[n*68 + k] = W[k*64 + n]
  for (int i = tid; i < HID * HID; i += 128) {
    const int k = i >> 6, n = i & 63;
    W1t[n * WT_STRIDE + k] = W1[i];
    W2t[n * WT_STRIDE + k] = W2[i];
  }
  if (tid < HID) {
    sb1[tid] = bias1[tid]; sb2[tid] = bias2[tid];
    ssum[tid] = 0.f; ssqs[tid] = 0.f;
  }
  __syncthreads();

  const int wave = tid >> 5;
  const int lane = tid & 31;
  const int M    = lane & 15;   // A: row within tile / B: col / C: N index
  const int kh   = lane >> 4;   // which K pair within a K=4 step
  const int rowBase = (blockIdx.x * 4 + wave) * 16;
  const float epsv  = 1.0f + epsp[0];

  // ---- A fragments for GEMM1: a_row = agg + (1+eps)*x, whole 16x64 in regs
  v2f a[16];
#pragma unroll
  for (int j = 0; j < 16; ++j) {
    const int row = rowBase + M;
    const int k   = j * 4 + kh * 2;
    v2f g = {0.f, 0.f}, xv = {0.f, 0.f};
    if (row < nNodes) {
      const size_t o = (size_t)row * HID + k;
      g  = *(const v2f*)(agg + o);
      xv = *(const v2f*)(xin + o);
    }
    a[j] = g + epsv * xv;
  }

  const v8f vzero = {0.f, 0.f, 0.f, 0.f, 0.f, 0.f, 0.f, 0.f};
  v8f c[4] = {vzero, vzero, vzero, vzero};

  // ---- GEMM1: (16x64) @ W1 -> 4 N-tiles of 16x16, K in 16 steps of 4
#pragma unroll
  for (int j = 0; j < 16; ++j) {
    const int k = j * 4 + kh * 2;
#pragma unroll
    for (int t = 0; t < 4; ++t) {
      const v2f b = *(const v2f*)&W1t[(t * 16 + M) * WT_STRIDE + k];
      c[t] = __builtin_amdgcn_wmma_f32_16x16x4_f32(
          false, a[j], false, b, (short)0, c[t], false, false);
    }
  }

  // ---- h1 = relu(c + b1); transpose C-layout -> A-layout via LDS
  float* T = tileT[wave];
#pragma unroll
  for (int t = 0; t < 4; ++t) {
    const float bb = sb1[t * 16 + M];
#pragma unroll
    for (int r = 0; r < 8; ++r) {
      float v = c[t][r] + bb;
      v = v > 0.f ? v : 0.f;
      T[(r + kh * 8) * WT_STRIDE + t * 16 + M] = v;
    }
  }
  __syncthreads();

#pragma unroll
  for (int j = 0; j < 16; ++j) {
    const int k = j * 4 + kh * 2;
    a[j] = *(const v2f*)&T[M * WT_STRIDE + k];
  }
#pragma unroll
  for (int t = 0; t < 4; ++t) c[t] = vzero;

  // ---- GEMM2
#pragma unroll
  for (int j = 0; j < 16; ++j) {
    const int k = j * 4 + kh * 2;
#pragma unroll
    for (int t = 0; t < 4; ++t) {
      const v2f b = *(const v2f*)&W2t[(t * 16 + M) * WT_STRIDE + k];
      c[t] = __builtin_amdgcn_wmma_f32_16x16x4_f32(
          false, a[j], false, b, (short)0, c[t], false, false);
    }
  }

  // ---- h2 = relu(c + b2); store pre-BN h; accumulate BN statistics
#pragma unroll
  for (int t = 0; t < 4; ++t) {
    const float bb = sb2[t * 16 + M];
    float ps = 0.f, pq = 0.f;
#pragma unroll
    for (int r = 0; r < 8; ++r) {
      const int row = rowBase + r + kh * 8;
      float v = c[t][r] + bb;
      v = v > 0.f ? v : 0.f;
      if (row < nNodes) {
        hpre[(size_t)row * HID + t * 16 + M] = v;
        ps += v;
        pq += v * v;
      }
    }
    atomicAdd(&ssum[t * 16 + M], ps);
    atomicAdd(&ssqs[t * 16 + M], pq);
  }
  __syncthreads();
  if (tid < HID) {
    atomicAdd(&stats[tid], (double)ssum[tid]);         // global_atomic_add_f64
    atomicAdd(&stats[HID + tid], (double)ssqs[tid]);
  }
}

// ---- BatchNorm apply: y = gamma*(h-mean)*rsqrt(var+1e-5)+beta
__global__ void k_bn(const float* __restrict__ hpre,
                     const double* __restrict__ stats,
                     const float* __restrict__ gamma,
                     const float* __restrict__ beta,
                     float* __restrict__ out, int nNodes) {
  int g = blockIdx.x * blockDim.x + threadIdx.x;
  if (g >= nNodes * HID) return;
  int f = g & (HID - 1);
  double inv  = 1.0 / (double)nNodes;
  double mean = stats[f] * inv;
  double var  = stats[HID + f] * inv - mean * mean;
  float rs = rsqrtf((float)var + 1e-5f);
  out[g] = gamma[f] * (hpre[g] - (float)mean) * rs + beta[f];
}

// ---- global mean pool (segment sum via f32 atomics) -----------------------
__global__ void k_pool(const float* __restrict__ y, const int* __restrict__ batch,
                       float* __restrict__ psum, float* __restrict__ pcnt,
                       int nNodes) {
  int g = blockIdx.x * blockDim.x + threadIdx.x;
  if (g >= nNodes * HID) return;
  int node = g >> 6, f = g & (HID - 1);
  int b = batch[node];
  atomicAdd(&psum[b * HID + f], y[g]);
  if (f == 0) atomicAdd(&pcnt[b], 1.0f);
}

__global__ void k_poolfin(const float* __restrict__ psum,
                          const float* __restrict__ pcnt,
                          float* __restrict__ hout) {
  int g = blockIdx.x * blockDim.x + threadIdx.x;
  if (g >= NUM_GRAPHS_C * HID) return;
  float c = pcnt[g >> 6];
  c = c > 1.0f ? c : 1.0f;
  hout[g] = psum[g] / c;
}

// ---- readout head ----------------------------------------------------------
__global__ void k_lin1(const float* __restrict__ h, const float* __restrict__ W,
                       const float* __restrict__ b, float* __restrict__ z) {
  int g = blockIdx.x * blockDim.x + threadIdx.x;
  if (g >= NUM_GRAPHS_C * HID) return;
  int row = g >> 6, n = g & (HID - 1);
  const float* hr = h + row * HID;
  float acc = b[n];
#pragma unroll 8
  for (int k = 0; k < HID; ++k) acc += hr[k] * W[k * HID + n];
  z[g] = acc > 0.f ? acc : 0.f;
}

__global__ void k_lin2(const float* __restrict__ z, const float* __restrict__ W,
                       const float* __restrict__ b, float* __restrict__ out) {
  int g = blockIdx.x * blockDim.x + threadIdx.x;
  if (g >= NUM_GRAPHS_C * NUM_CLASSES) return;
  int row = g / NUM_CLASSES, n = g % NUM_CLASSES;
  const float* zr = z + row * HID;
  float acc = b[n];
#pragma unroll 8
  for (int k = 0; k < HID; ++k) acc += zr[k] * W[k * NUM_CLASSES + n];
  out[g] = acc;
}

// ---------------------------------------------------------------------------

extern "C" void kernel_launch(void* const* d_in, const int* in_sizes, int n_in,
                              void* d_out, int out_size, void* d_ws, size_t ws_size,
                              hipStream_t stream) {
  (void)in_sizes; (void)n_in; (void)out_size; (void)ws_size;

  const float* x    = (const float*)d_in[0];
  const int* esrc   = (const int*)d_in[1];
  const int* edst   = esrc + N_EDGES_C;
  const int* batch  = (const int*)d_in[2];

  const float *W1[3], *W2[3], *b1[3], *b2[3], *beta[3], *eps[3], *gamma[3];
  for (int l = 0; l < 3; ++l) {
    int base = 3 + 7 * l;  // sorted: W1, W2, b1, b2, beta, eps, gamma
    W1[l]    = (const float*)d_in[base + 0];
    W2[l]    = (const float*)d_in[base + 1];
    b1[l]    = (const float*)d_in[base + 2];
    b2[l]    = (const float*)d_in[base + 3];
    beta[l]  = (const float*)d_in[base + 4];
    eps[l]   = (const float*)d_in[base + 5];
    gamma[l] = (const float*)d_in[base + 6];
  }
  const float* lin1W = (const float*)d_in[24];
  const float* lin1b = (const float*)d_in[25];
  const float* lin2W = (const float*)d_in[26];
  const float* lin2b = (const float*)d_in[27];

  // workspace carve-up
  const size_t FEAT = (size_t)N_NODES_C * HID * sizeof(float);  // 25.6 MB
  char* ws = (char*)d_ws;
  float*  agg   = (float*)(ws + 0 * FEAT);
  float*  hpre  = (float*)(ws + 1 * FEAT);
  float*  bufA  = (float*)(ws + 2 * FEAT);
  float*  bufB  = (float*)(ws + 3 * FEAT);
  double* stats = (double*)(ws + 4 * FEAT);            // 128 doubles
  float*  psum  = (float*)(ws + 4 * FEAT + 1024);      // 256*64 floats
  float*  pcnt  = psum + NUM_GRAPHS_C * HID;           // 256 floats (contig.)
  float*  zbuf  = pcnt + NUM_GRAPHS_C;                 // 256*64 floats

  const int NF = N_NODES_C * HID;  // 6,400,000

  const float* cur = x;
  for (int l = 0; l < 3; ++l) {
    k_zero<<<(NF + 255) / 256, 256, 0, stream>>>(agg, NF);
    k_zero<<<1, 256, 0, stream>>>((float*)stats, 256);  // 128 doubles
    k_scatter<<<(N_EDGES_C * 16 + 255) / 256, 256, 0, stream>>>(
        cur, esrc, edst, agg, N_EDGES_C);
    k_mlp<<<(N_NODES_C + 63) / 64, 128, 0, stream>>>(
        cur, agg, W1[l], b1[l], W2[l], b2[l], eps[l], hpre, stats, N_NODES_C);
    float* y = (l == 0) ? bufA : (l == 1) ? bufB : bufA;
    k_bn<<<(NF + 255) / 256, 256, 0, stream>>>(hpre, stats, gamma[l], beta[l],
                                               y, N_NODES_C);
    cur = y;
  }

  // pooled mean -> first 256*64 elements of d_out
  k_zero<<<((NUM_GRAPHS_C * HID + NUM_GRAPHS_C) + 255) / 256, 256, 0, stream>>>(
      psum, NUM_GRAPHS_C * HID + NUM_GRAPHS_C);
  k_pool<<<(NF + 255) / 256, 256, 0, stream>>>(cur, batch, psum, pcnt, N_NODES_C);
  float* out_h = (float*)d_out;
  k_poolfin<<<(NUM_GRAPHS_C * HID + 255) / 256, 256, 0, stream>>>(psum, pcnt, out_h);

  // head: z = relu(h@lin1+b); logits = z@lin2+b -> d_out[16384..]
  k_lin1<<<(NUM_GRAPHS_C * HID + 255) / 256, 256, 0, stream>>>(out_h, lin1W, lin1b, zbuf);
  k_lin2<<<(NUM_GRAPHS_C * NUM_CLASSES + 255) / 256, 256, 0, stream>>>(
      zbuf, lin2W, lin2b, out_h + NUM_GRAPHS_C * HID);
}